// GCN1_80444737454321
// MI455X (gfx1250) — compile-verified
//
#include <hip/hip_runtime.h>
#include <hip/hip_bf16.h>
#include <stdint.h>

typedef __attribute__((ext_vector_type(2))) float v2f;
typedef __attribute__((ext_vector_type(8))) float v8f;

#define N_NODES 100000
#define N_EDGES 1600000

// ---------------------------------------------------------------------------
// Degree / normalization
// ---------------------------------------------------------------------------
__global__ void gcn_deg_init(float* deg, int n) {
    int i = blockIdx.x * blockDim.x + threadIdx.x;
    if (i < n) deg[i] = 1.0f;  // self-loop contribution
}

__global__ void gcn_deg_accum(const int64_t* __restrict__ ei, float* deg, int E) {
    int e = blockIdx.x * blockDim.x + threadIdx.x;
    if (e < E) atomicAdd(&deg[(int)ei[(int64_t)E + e]], 1.0f);
}

__global__ void gcn_make_dis(float* deg, int n) {
    int i = blockIdx.x * blockDim.x + threadIdx.x;
    if (i < n) deg[i] = rsqrtf(deg[i]);  // deg >= 1 always (self loop)
}

// ---------------------------------------------------------------------------
// WMMA f32 GEMM: T[M x NOUT] = (relu?)A[M x K] @ W[K x NOUT]
// One wave computes a 16-row strip x all NOUT columns (NTILES 16-col tiles).
// V_WMMA_F32_16X16X4_F32: A 16x4, B 4x16, C/D 16x16 f32.
// A-frag (32-bit A 16x4): lanes 0-15 -> K=k,k+1 ; lanes 16-31 -> K=k+2,k+3.
// B-frag: VGPR0 rows K=k (lanes0-15) / K=k+2 (lanes16-31); VGPR1 K=k+1/K=k+3.
// D: VGPR v -> M = v + 8*(lane>>4), N = lane&15.
// NOTE: nrows must be a multiple of 16 (true here: 100000 = 6250*16), so the
// store epilogue carries no per-row guard (keeps EXEC uniform, no branch
// cascade). Only a tile that straddles NOUT (layer-4, cols 32..47 vs NOUT=40)
// keeps a per-column guard; full tiles fold it away at unroll time.
// ---------------------------------------------------------------------------
template<int K, int NOUT, int NTILES, bool RELU_IN>
__global__ void gcn_gemm_wmma(const float* __restrict__ A,
                              const float* __restrict__ W,
                              float* __restrict__ T, int nrows) {
    int wave  = (blockIdx.x * blockDim.x + threadIdx.x) >> 5;
    int strip = wave;
    if (strip * 16 >= nrows) return;  // uniform per wave -> EXEC stays all-1s

    const int lane = threadIdx.x & 31;
    const int half = lane >> 4;   // 0 or 1
    const int l15  = lane & 15;
    const int koff = half * 2;
    const int arow = strip * 16 + l15;

    v8f acc[NTILES];
#pragma unroll
    for (int t = 0; t < NTILES; ++t) {
        v8f z = {0.f, 0.f, 0.f, 0.f, 0.f, 0.f, 0.f, 0.f};
        acc[t] = z;
    }

    for (int k = 0; k < K; k += 4) {
        float a0 = A[(int64_t)arow * K + k + koff];
        float a1 = A[(int64_t)arow * K + k + koff + 1];
        if (RELU_IN) { a0 = fmaxf(a0, 0.0f); a1 = fmaxf(a1, 0.0f); }
        v2f a; a.x = a0; a.y = a1;
#pragma unroll
        for (int t = 0; t < NTILES; ++t) {
            const bool full_tile = (t + 1) * 16 <= NOUT;  // folds after unroll
            int col = t * 16 + l15;
            v2f b;
            if (full_tile) {
                b.x = W[(k + koff) * NOUT + col];
                b.y = W[(k + koff + 1) * NOUT + col];
            } else {
                b.x = (col < NOUT) ? W[(k + koff) * NOUT + col] : 0.0f;
                b.y = (col < NOUT) ? W[(k + koff + 1) * NOUT + col] : 0.0f;
            }
            acc[t] = __builtin_amdgcn_wmma_f32_16x16x4_f32(
                false, a, false, b, (short)0, acc[t], false, false);
        }
    }

#pragma unroll
    for (int t = 0; t < NTILES; ++t) {
        const bool full_tile = (t + 1) * 16 <= NOUT;  // folds after unroll
        int col = t * 16 + l15;
#pragma unroll
        for (int v = 0; v < 8; ++v) {
            int r = strip * 16 + v + 8 * half;
            if (full_tile || col < NOUT)
                T[(int64_t)r * NOUT + col] = acc[t][v];
        }
    }
}

// ---------------------------------------------------------------------------
// Aggregation: S[i] = b + dis[i]^2 * T[i]  (self-loop + bias, no atomics)
// ---------------------------------------------------------------------------
template<int NOUT>
__global__ void gcn_agg_init(const float* __restrict__ T,
                             const float* __restrict__ bias,
                             const float* __restrict__ dis,
                             float* __restrict__ S, int n) {
    unsigned gid = blockIdx.x * blockDim.x + threadIdx.x;
    if (gid >= (unsigned)n * NOUT) return;
    int i = gid / NOUT;
    int c = gid % NOUT;
    float d = dis[i];
    S[gid] = bias[c] + d * d * T[gid];
}

// ---------------------------------------------------------------------------
// Edge scatter: S[dst] += dis[src]*dis[dst] * T[src], float4-vectorized,
// resolves in L2 (both 51MB feature buffers fit in the 192MB L2).
// ---------------------------------------------------------------------------
template<int NOUT>
__global__ void gcn_agg_scatter(const float* __restrict__ T,
                                const int64_t* __restrict__ ei,
                                const float* __restrict__ dis,
                                float* __restrict__ S, int E) {
    const int TPE = NOUT / 4;  // float4 chunks per edge
    unsigned gid = blockIdx.x * blockDim.x + threadIdx.x;
    if (gid >= (unsigned)E * TPE) return;
    int e = gid / TPE;
    int q = gid % TPE;
    int src = (int)ei[e];
    int dst = (int)ei[(int64_t)E + e];
    float w = dis[src] * dis[dst];
    const float4 v = *(const float4*)(T + (int64_t)src * NOUT + q * 4);
    float* sp = S + (int64_t)dst * NOUT + q * 4;
    atomicAdd(sp + 0, w * v.x);
    atomicAdd(sp + 1, w * v.y);
    atomicAdd(sp + 2, w * v.z);
    atomicAdd(sp + 3, w * v.w);
}

// ---------------------------------------------------------------------------
// log_softmax over 40 classes, one thread per node (16MB, L2-resident)
// ---------------------------------------------------------------------------
__global__ void gcn_log_softmax40(const float* __restrict__ S,
                                  float* __restrict__ out, int n) {
    int i = blockIdx.x * blockDim.x + threadIdx.x;
    if (i >= n) return;
    const float* r = S + (int64_t)i * 40;
    float m = r[0];
#pragma unroll
    for (int c = 1; c < 40; ++c) m = fmaxf(m, r[c]);
    float sum = 0.0f;
#pragma unroll
    for (int c = 0; c < 40; ++c) sum += __expf(r[c] - m);
    float ls = __logf(sum);
    float* o = out + (int64_t)i * 40;
#pragma unroll
    for (int c = 0; c < 40; ++c) o[c] = r[c] - m - ls;
}

// ---------------------------------------------------------------------------
// Host orchestration
// ---------------------------------------------------------------------------
static inline int ceil_div(int a, int b) { return (a + b - 1) / b; }

extern "C" void kernel_launch(void* const* d_in, const int* in_sizes, int n_in,
                              void* d_out, int out_size, void* d_ws, size_t ws_size,
                              hipStream_t stream) {
    const float*   x  = (const float*)d_in[0];
    const int64_t* ei = (const int64_t*)d_in[1];
    const float*   W1 = (const float*)d_in[2];
    const float*   b1 = (const float*)d_in[3];
    const float*   W2 = (const float*)d_in[4];
    const float*   b2 = (const float*)d_in[5];
    const float*   W3 = (const float*)d_in[6];
    const float*   b3 = (const float*)d_in[7];
    const float*   W4 = (const float*)d_in[8];
    const float*   b4 = (const float*)d_in[9];
    float* out = (float*)d_out;

    const int n = N_NODES, E = N_EDGES;

    // workspace layout: dis (n floats) | bufA (n*128) | bufB (n*128)
    char* ws = (char*)d_ws;
    float* dis  = (float*)ws;
    float* bufA = (float*)(ws + (size_t)n * sizeof(float));
    float* bufB = bufA + (size_t)n * 128;

    const int TB = 256;
    const int strips = n / 16;  // n is a multiple of 16 (100000 = 6250*16)
    const int gemm_blocks = ceil_div(strips * 32, TB);

    // degree -> dis
    gcn_deg_init<<<ceil_div(n, TB), TB, 0, stream>>>(dis, n);
    gcn_deg_accum<<<ceil_div(E, TB), TB, 0, stream>>>(ei, dis, E);
    gcn_make_dis<<<ceil_div(n, TB), TB, 0, stream>>>(dis, n);

    // Layer 1: x[100k x 256] @ W1[256 x 64]
    gcn_gemm_wmma<256, 64, 4, false><<<gemm_blocks, TB, 0, stream>>>(x, W1, bufA, n);
    gcn_agg_init<64><<<ceil_div(n * 64, TB), TB, 0, stream>>>(bufA, b1, dis, bufB, n);
    gcn_agg_scatter<64><<<ceil_div(E * 16, TB), TB, 0, stream>>>(bufA, ei, dis, bufB, E);

    // Layer 2: relu(s1)[100k x 64] @ W2[64 x 128]
    gcn_gemm_wmma<64, 128, 8, true><<<gemm_blocks, TB, 0, stream>>>(bufB, W2, bufA, n);
    gcn_agg_init<128><<<ceil_div(n * 128, TB), TB, 0, stream>>>(bufA, b2, dis, bufB, n);
    gcn_agg_scatter<128><<<ceil_div(E * 32, TB), TB, 0, stream>>>(bufA, ei, dis, bufB, E);

    // Layer 3: relu(s2)[100k x 128] @ W3[128 x 64]
    gcn_gemm_wmma<128, 64, 4, true><<<gemm_blocks, TB, 0, stream>>>(bufB, W3, bufA, n);
    gcn_agg_init<64><<<ceil_div(n * 64, TB), TB, 0, stream>>>(bufA, b3, dis, bufB, n);
    gcn_agg_scatter<64><<<ceil_div(E * 16, TB), TB, 0, stream>>>(bufA, ei, dis, bufB, E);

    // Layer 4: relu(s3)[100k x 64] @ W4[64 x 40] (3 tiles, last one partial)
    gcn_gemm_wmma<64, 40, 3, true><<<gemm_blocks, TB, 0, stream>>>(bufB, W4, bufA, n);
    gcn_agg_init<40><<<ceil_div(n * 40, TB), TB, 0, stream>>>(bufA, b4, dis, bufB, n);
    gcn_agg_scatter<40><<<ceil_div(E * 10, TB), TB, 0, stream>>>(bufA, ei, dis, bufB, E);

    // log_softmax -> d_out
    gcn_log_softmax40<<<ceil_div(n, TB), TB, 0, stream>>>(bufB, out, n);
}